// RecursiveNet_4990751998296
// MI455X (gfx1250) — compile-verified
//
#include <hip/hip_runtime.h>

// ---------------------------------------------------------------------------
// RecursiveNet on MI455X (gfx1250): bottom-up tree eval as per-symbol GEMMs
// using V_WMMA_F32_16X16X4_F32 (fp32-accurate, workload is latency-bound),
// with async global->LDS staging (GLOBAL_LOAD_ASYNC_TO_LDS_B128) of W_s.
// ---------------------------------------------------------------------------

#define DIM      64
#define OUT_DIM  32
#define ALPHA    16
#define N_NODES  21845
#define WPAD     66   // padded LDS row stride (floats) to avoid bank conflicts

typedef float v2f __attribute__((ext_vector_type(2)));
typedef float v8f __attribute__((ext_vector_type(8)));
typedef int   v4i_vs __attribute__((vector_size(16)));   // matches builtin param

#if __has_builtin(__builtin_amdgcn_global_load_async_to_lds_b128) && \
    __has_builtin(__builtin_amdgcn_s_wait_asynccnt)
#define HAVE_ASYNC_LDS 1
#else
#define HAVE_ASYNC_LDS 0
#endif

__device__ __forceinline__ float sigmoid_f(float x) {
    // v_exp_f32 + v_rcp_f32 instead of the full IEEE divide expansion.
    return __builtin_amdgcn_rcpf(1.0f + __expf(-x));
}

// --------------------------- leaves: e = sigma(b[sym]) ----------------------
__global__ void rn_leaf_kernel(const int* __restrict__ sym,
                               const float* __restrict__ b,
                               float* __restrict__ enc,
                               int off, int size) {
    int i = blockIdx.x * blockDim.x + threadIdx.x;
    if (i >= size * DIM) return;
    int n = i / DIM;
    int d = i - n * DIM;
    int s = sym[off + n];
    enc[(size_t)(off + n) * DIM + d] = sigmoid_f(b[s * DIM + d]);
}

// --------------- per-level counting sort of nodes by symbol -----------------
// seg[0..15] = segment start, seg[16..31] = segment count (within level)
__global__ void rn_bucket_kernel(const int* __restrict__ sym,
                                 int off, int size,
                                 int* __restrict__ order,
                                 int* __restrict__ seg) {
    __shared__ int cnt[ALPHA];
    __shared__ int cur[ALPHA];
    int t = threadIdx.x;
    if (t < ALPHA) cnt[t] = 0;
    __syncthreads();
    for (int i = t; i < size; i += blockDim.x)
        atomicAdd(&cnt[sym[off + i]], 1);
    __syncthreads();
    if (t == 0) {
        int run = 0;
        for (int s = 0; s < ALPHA; ++s) {
            cur[s] = run;
            seg[s] = run;
            seg[ALPHA + s] = cnt[s];
            run += cnt[s];
        }
    }
    __syncthreads();
    for (int i = t; i < size; i += blockDim.x) {
        int s = sym[off + i];
        int pos = atomicAdd(&cur[s], 1);
        order[pos] = off + i;   // global node id, symbol-sorted within level
    }
}

// ---------------- per-level GEMM tile: 16 nodes x one symbol ----------------
// grid = (ceil(size/16), ALPHA), block = 32 (one wave32 per tile)
__global__ void __launch_bounds__(32)
rn_level_gemm_kernel(const int* __restrict__ children,
                     const float* __restrict__ W,
                     const float* __restrict__ b,
                     const int* __restrict__ order,
                     const int* __restrict__ seg,
                     float* __restrict__ enc) {
    const int s = blockIdx.y;
    const int start = seg[s];
    const int count = seg[ALPHA + s];
    const int r0 = blockIdx.x * 16;
    if (r0 >= count) return;                       // wave-uniform exit
    const int valid = (count - r0 < 16) ? (count - r0) : 16;

    __shared__ float Ws[DIM * WPAD];               // staged W[s], padded
    __shared__ float Hs[16 * WPAD];                // H tile (16 x 64), padded
    __shared__ float bs[DIM];
    __shared__ int   nid[16];

    const int lane = threadIdx.x;                  // 0..31
    const float* __restrict__ Wsrc = W + (size_t)s * DIM * DIM;
    __builtin_prefetch(Wsrc, 0, 3);                // global_prefetch_b8

    // ---- kick off async W_s staging first (ASYNCcnt), overlap with H build
#if HAVE_ASYNC_LDS
    for (int i = lane; i < (DIM * DIM) / 4; i += 32) {   // 1024 x 16B chunks
        const int r  = i >> 4;                           // row (64 rows)
        const int c4 = (i & 15) * 4;                     // col (floats)
        __builtin_amdgcn_global_load_async_to_lds_b128(
            (__attribute__((address_space(1))) v4i_vs*)(Wsrc + r * DIM + c4),
            (__attribute__((address_space(3))) v4i_vs*)(&Ws[r * WPAD + c4]),
            0, 0);
    }
#else
    for (int i = lane; i < DIM * DIM; i += 32) {
        int r = i >> 6, c = i & 63;
        Ws[r * WPAD + c] = Wsrc[i];
    }
#endif

    if (lane < 16)
        nid[lane] = (lane < valid) ? order[start + r0 + lane] : -1;
    for (int i = lane; i < DIM; i += 32)
        bs[i] = b[s * DIM + i];
    __syncthreads();

    // Build H: h = mean of 4 children's encodings (2 dims per lane, per row).
    // This overlaps with the in-flight async W copy.
    const int d = lane * 2;
    for (int r = 0; r < 16; ++r) {
        float h0 = 0.0f, h1 = 0.0f;
        int n = nid[r];
        if (n >= 0) {
            const int4 ch = *(const int4*)(children + 4 * n);
            const float2 a = *(const float2*)(enc + (size_t)ch.x * DIM + d);
            const float2 bb = *(const float2*)(enc + (size_t)ch.y * DIM + d);
            const float2 c = *(const float2*)(enc + (size_t)ch.z * DIM + d);
            const float2 e = *(const float2*)(enc + (size_t)ch.w * DIM + d);
            h0 = (a.x + bb.x + c.x + e.x) * 0.25f;
            h1 = (a.y + bb.y + c.y + e.y) * 0.25f;
        }
        Hs[r * WPAD + d]     = h0;
        Hs[r * WPAD + d + 1] = h1;
    }
#if HAVE_ASYNC_LDS
    __builtin_amdgcn_s_wait_asynccnt(0);           // W_s landed in LDS
#endif
    __syncthreads();

    // E(16x64) = sigmoid( H(16x64) @ W_s^T + b_s ), via 16x16x4 f32 WMMA.
    // A frag (16x4): lanes 0-15 -> K=0,1 ; lanes 16-31 -> K=2,3 (rows = lane%16)
    // B frag (4x16): B[kk][nn] = W_s[n0+nn][k0+kk], same lane split on K.
    const int row = lane & 15;
    const int hi  = lane >> 4;
#pragma unroll
    for (int nt = 0; nt < 4; ++nt) {
        v8f acc = {};
#pragma unroll
        for (int kk = 0; kk < 16; ++kk) {
            const int k0 = 4 * kk;
            const int koff = k0 + 2 * hi;
            v2f afr, bfr;
            afr.x = Hs[row * WPAD + koff];
            afr.y = Hs[row * WPAD + koff + 1];
            const int wr = nt * 16 + row;
            bfr.x = Ws[wr * WPAD + koff];
            bfr.y = Ws[wr * WPAD + koff + 1];
            acc = __builtin_amdgcn_wmma_f32_16x16x4_f32(
                false, afr, false, bfr, (short)0, acc, false, false);
        }
        // D layout: VGPR v -> row v (lanes 0-15) / row v+8 (lanes 16-31)
        const int col = nt * 16 + row;
        const float bb = bs[col];
#pragma unroll
        for (int v = 0; v < 8; ++v) {
            const int drow = v + 8 * hi;
            if (drow < valid) {
                const int n = nid[drow];
                enc[(size_t)n * DIM + col] = sigmoid_f(acc[v] + bb);
            }
        }
    }
}

// -------------------------- out = enc[0] @ W_out^T + b_out ------------------
__global__ void rn_out_kernel(const float* __restrict__ enc,
                              const float* __restrict__ W_out,
                              const float* __restrict__ b_out,
                              float* __restrict__ out) {
    int lane = threadIdx.x;            // 32 lanes == OUT_DIM
    float acc = b_out[lane];
#pragma unroll 8
    for (int k = 0; k < DIM; ++k)
        acc += enc[k] * W_out[lane * DIM + k];
    out[lane] = acc;
}

// ---------------------------------------------------------------------------
extern "C" void kernel_launch(void* const* d_in, const int* in_sizes, int n_in,
                              void* d_out, int out_size, void* d_ws, size_t ws_size,
                              hipStream_t stream) {
    (void)in_sizes; (void)n_in; (void)out_size; (void)ws_size;
    const int*   sym      = (const int*)d_in[0];
    const int*   children = (const int*)d_in[1];
    const float* W        = (const float*)d_in[2];
    const float* b        = (const float*)d_in[3];
    const float* W_out    = (const float*)d_in[4];
    const float* b_out    = (const float*)d_in[5];
    float*       out      = (float*)d_out;

    float* enc  = (float*)d_ws;                          // N*64 f32
    int* order  = (int*)(enc + (size_t)N_NODES * DIM);   // N ints
    int* seg    = order + N_NODES;                       // 7 levels * 32 ints

    static const int OFFS[9] = {0, 1, 5, 21, 85, 341, 1365, 5461, 21845};

    // Level 7 (leaves): e = sigmoid(b[sym]) — no matvec needed.
    {
        const int off = OFFS[7], size = OFFS[8] - OFFS[7];
        const int total = size * DIM;
        rn_leaf_kernel<<<(total + 255) / 256, 256, 0, stream>>>(sym, b, enc, off, size);
    }

    // Levels 6..0: bucket by symbol, then per-symbol GEMM tiles.
    for (int l = 6; l >= 0; --l) {
        const int off = OFFS[l], size = OFFS[l + 1] - OFFS[l];
        rn_bucket_kernel<<<1, 256, 0, stream>>>(sym, off, size, order + off, seg + l * 32);
        const int tiles = (size + 15) / 16;
        dim3 grid(tiles, ALPHA);
        rn_level_gemm_kernel<<<grid, 32, 0, stream>>>(children, W, b,
                                                      order + off, seg + l * 32, enc);
    }

    rn_out_kernel<<<1, 32, 0, stream>>>(enc, W_out, b_out, out);
}